// Encoder_34368328303098
// MI455X (gfx1250) — compile-verified
//
#include <hip/hip_runtime.h>

typedef __attribute__((ext_vector_type(2))) float        v2f;
typedef __attribute__((ext_vector_type(8))) float        v8f;
typedef __attribute__((ext_vector_type(4))) unsigned int u32x4;
typedef __attribute__((ext_vector_type(4))) int          i32x4;
typedef __attribute__((ext_vector_type(8))) int          i32x8;

#define NA 10000
#define NB 20000
#define DIM 128
#define NJT (NB / 16)                    // 1250 B tiles of 16 rows
#define GROUPS 8                         // A tiles per block
#define PHASES 4                         // B-sweep split
#define ROWS_PER_BLOCK (GROUPS * 16)     // 128
#define STAGE_TILES PHASES               // B tiles staged per stage
#define STAGE_ROWS (STAGE_TILES * 16)    // 64 rows = 32 KB
#define NSTAGES ((NJT + STAGE_TILES - 1) / STAGE_TILES)  // 313 (last stage: 2 tiles)

// ---- Pass 1: column norms ||b_j||^2 --------------------------------------
__global__ __launch_bounds__(256) void b2_kernel(const float* __restrict__ B,
                                                 float* __restrict__ b2) {
  int j = blockIdx.x * blockDim.x + threadIdx.x;
  if (j >= NB) return;
  const float4* row = (const float4*)(B + (size_t)j * DIM);
  float s = 0.f;
#pragma unroll
  for (int k = 0; k < DIM / 4; ++k) {
    float4 v = row[k];
    s += v.x * v.x + v.y * v.y + v.z * v.z + v.w * v.w;
  }
  b2[j] = s;
}

// Issue one TDM 2D tile DMA: rows x 128 f32, global (row-major, stride 128)
// -> LDS at ldsByteAddr. D# packing per CDNA5 ISA §8.3/8.4.
__device__ __forceinline__ void tdm_stage(unsigned int ldsByteAddr,
                                          const float* __restrict__ gsrc,
                                          int rows) {
  unsigned long long ga = (unsigned long long)(uintptr_t)gsrc;
  u32x4 g0 = { 1u,                                    // count=1, user mode
               ldsByteAddr,                           // lds_addr
               (unsigned int)(ga & 0xffffffffu),      // global_addr[31:0]
               (unsigned int)((ga >> 32) & 0x1ffffffu) | (2u << 30) }; // [56:32] | type=2
  i32x8 g1 = { (int)(2u << 16),                       // data_size = 4 bytes
               (int)(128u << 16),                     // tensor_dim0 = 128 (bits 79:48 lo)
               (int)(((unsigned)rows) << 16),         // tensor_dim1 = rows (bits 111:80 lo)
               (int)(128u << 16),                     // tile_dim0 = 128 (bits 127:112)
               rows,                                  // tile_dim1 = rows (bits 143:128)
               128,                                   // tensor_dim0_stride = 128 (bits 191:160)
               0, 0 };
  i32x4 z4 = { 0, 0, 0, 0 };
#if __has_include(<hip/amd_detail/amd_gfx1250_TDM.h>)
  i32x8 z8 = { 0, 0, 0, 0, 0, 0, 0, 0 };
  __builtin_amdgcn_tensor_load_to_lds(g0, g1, z4, z4, z8, 0);   // 6-arg toolchain
#else
  __builtin_amdgcn_tensor_load_to_lds(g0, g1, z4, z4, 0);       // ROCm 7.2, 5-arg
#endif
}

// ---- Pass 2: argmin_j (||b_j||^2 - 2 a_i.b_j) via f32 WMMA + TDM staging --
__global__ __launch_bounds__(1024) void nn_kernel(const float* __restrict__ A,
                                                  const float* __restrict__ B,
                                                  const float* __restrict__ b2,
                                                  float* __restrict__ out) {
  const int tid  = threadIdx.x;
  const int lane = tid & 31;
  const int wave = tid >> 5;      // 0..31
  const int g    = wave >> 2;     // A-tile group 0..7
  const int p    = wave & 3;      // B-sweep phase 0..3

  __shared__ float sB[2][STAGE_ROWS * DIM];          // 2 x 32 KB staging
  __shared__ float sScore[GROUPS][16][PHASES];
  __shared__ int   sIdx[GROUPS][16][PHASES];
  __shared__ int   sBest[GROUPS][16];

  // WMMA f32 16x16x4 A-layout: lanes 0-15: M=lane, K={4t,4t+1};
  //                            lanes 16-31: M=lane-16, K={4t+2,4t+3}.
  // B fragment(K=k, N=n) = Brow_n[k]: transpose-symmetric, same per-lane
  // float2 addressing -> D[m][n] = dot(A_row_m, B_row_n).
  const int m    = lane & 15;
  const int koff = (lane >> 4) * 2;

  const int aRow0 = blockIdx.x * ROWS_PER_BLOCK + g * 16;
  const int aRowC = aRow0 <= NA - 16 ? aRow0 : NA - 16;   // clamp tail (dup work, writes guarded)

  v2f aF[32];
  const float* aPtr = A + (size_t)(aRowC + m) * DIM + koff;
#pragma unroll
  for (int t = 0; t < 32; ++t)
    aF[t] = *(const v2f*)(aPtr + 4 * t);

  float bestS[8];
  int   bestI[8];
#pragma unroll
  for (int r = 0; r < 8; ++r) { bestS[r] = __builtin_inff(); bestI[r] = 0x7fffffff; }

  // Prologue: stage 0 into buffer 0.
  if (wave == 0) {
    tdm_stage((unsigned int)(uintptr_t)&sB[0][0], B, STAGE_ROWS);
    __builtin_amdgcn_s_wait_tensorcnt((short)0);
  }
  __syncthreads();

  for (int s = 0; s < NSTAGES; ++s) {
    const int cur = s & 1;
    const int nxt = cur ^ 1;
    const bool more = (s + 1) < NSTAGES;

    // Kick DMA for the next stage; overlaps this stage's WMMA chains.
    if (wave == 0 && more) {
      int nrows = NB - (s + 1) * STAGE_ROWS;
      nrows = nrows < STAGE_ROWS ? nrows : STAGE_ROWS;
      tdm_stage((unsigned int)(uintptr_t)&sB[nxt][0],
                B + (size_t)(s + 1) * STAGE_ROWS * DIM, nrows);
    }

    const int jt = s * STAGE_TILES + p;
    if (jt < NJT) {                       // wave-uniform; last stage phases 2,3 skip
      const int rowBase = (p * 16 + m) * DIM + koff;
      v8f c = {};
#pragma unroll
      for (int t = 0; t < 32; ++t) {
        v2f bF = *(const v2f*)&sB[cur][rowBase + 4 * t];   // ds_load_b64
        // 8 args: (neg_a, A, neg_b, B, c_mod, C, reuse_a, reuse_b)
        c = __builtin_amdgcn_wmma_f32_16x16x4_f32(false, aF[t], false, bF,
                                                  (short)0, c, false, false);
      }
      const float bn   = b2[jt * 16 + m];
      const int   jcol = jt * 16 + m;
#pragma unroll
      for (int r = 0; r < 8; ++r) {
        // C/D layout: component r, lanes 0-15 -> row r; lanes 16-31 -> row r+8.
        float sc = bn - 2.0f * c[r];
        if (sc < bestS[r]) { bestS[r] = sc; bestI[r] = jcol; }  // first min wins
      }
    }

    if (wave == 0 && more) __builtin_amdgcn_s_wait_tensorcnt((short)0);
    __syncthreads();    // next buffer ready for everyone; old buffer free
  }

  // Cross-lane argmin within each 16-lane half (ties -> smaller index).
#pragma unroll
  for (int r = 0; r < 8; ++r) {
    float s = bestS[r];
    int   i = bestI[r];
#pragma unroll
    for (int mask = 8; mask >= 1; mask >>= 1) {
      float os = __shfl_xor(s, mask, 32);
      int   oi = __shfl_xor(i, mask, 32);
      if (os < s || (os == s && oi < i)) { s = os; i = oi; }
    }
    bestS[r] = s; bestI[r] = i;
  }

  if (lane == 0) {
#pragma unroll
    for (int r = 0; r < 8; ++r) { sScore[g][r][p] = bestS[r]; sIdx[g][r][p] = bestI[r]; }
  } else if (lane == 16) {
#pragma unroll
    for (int r = 0; r < 8; ++r) { sScore[g][r + 8][p] = bestS[r]; sIdx[g][r + 8][p] = bestI[r]; }
  }
  __syncthreads();

  // Combine the 4 sweep phases per (group, row).
  if (tid < GROUPS * 16) {
    const int gg = tid >> 4, row = tid & 15;
    float s = sScore[gg][row][0];
    int   i = sIdx[gg][row][0];
#pragma unroll
    for (int w = 1; w < PHASES; ++w) {
      float os = sScore[gg][row][w];
      int   oi = sIdx[gg][row][w];
      if (os < s || (os == s && oi < i)) { s = os; i = oi; }
    }
    sBest[gg][row] = i;
  }
  __syncthreads();

  // Gather: 128 output rows x 512 B; 1024 threads x 64 B each.
  const int orow = tid >> 3;            // 0..127
  const int ch   = tid & 7;             // 16 floats per thread
  const int aRowOut = blockIdx.x * ROWS_PER_BLOCK + orow;
  if (aRowOut < NA) {
    const int bi = sBest[orow >> 4][orow & 15];
    const float4* src = (const float4*)(B + (size_t)bi * DIM + ch * 16);
    float4*       dst = (float4*)(out + (size_t)aRowOut * DIM + ch * 16);
#pragma unroll
    for (int q = 0; q < 4; ++q) dst[q] = src[q];
  }
}

extern "C" void kernel_launch(void* const* d_in, const int* in_sizes, int n_in,
                              void* d_out, int out_size, void* d_ws, size_t ws_size,
                              hipStream_t stream) {
  const float* A = (const float*)d_in[0];   // embeddingA [10000,128] f32
  const float* B = (const float*)d_in[1];   // embeddingB [20000,128] f32
  float* out = (float*)d_out;               // [10000,128] f32
  float* b2  = (float*)d_ws;                // 20000 floats = 80 KB scratch

  b2_kernel<<<(NB + 255) / 256, 256, 0, stream>>>(B, b2);
  const int nblocks = (NA + ROWS_PER_BLOCK - 1) / ROWS_PER_BLOCK;   // 79
  nn_kernel<<<nblocks, 1024, 0, stream>>>(A, B, b2, out);
}